// Polar2Cart_7043746365525
// MI455X (gfx1250) — compile-verified
//
#include <hip/hip_runtime.h>

#define H_  512
#define W_  512
#define HP_ 64
#define WP_ 2048
#define B_  4
#define C_  64
#define PPLANE_ (HP_ * WP_)   // polar channel plane:  131072 elems (512 KB)
#define OPLANE_ (H_ * W_)     // output channel plane: 262144 elems (1 MB)

typedef float v4f __attribute__((ext_vector_type(4)));

// ---------------------------------------------------------------------------
// Kernel: scatter sample index n into map[idx_y[n]*W + idx_x[n]]
// (masked pixels are unique, so no write conflicts; map pre-set to -1 by memset)
// ---------------------------------------------------------------------------
__global__ __launch_bounds__(256) void p2c_build_map(const int* __restrict__ iyv,
                                                     const int* __restrict__ ixv,
                                                     int N, int* __restrict__ map) {
    int n = blockIdx.x * 256 + threadIdx.x;
    if (n < N) map[iyv[n] * W_ + ixv[n]] = n;
}

// ---------------------------------------------------------------------------
// Bilinear tap setup. Replicates the reference's
// grid_sample(mode='bilinear', padding_mode='zeros', align_corners=True).
// ---------------------------------------------------------------------------
struct Taps {
    int   o00, o01, o10, o11;   // element offsets into one [HP,WP] plane
    float w00, w01, w10, w11;   // weights (0 where out of bounds)
};

__device__ __forceinline__ Taps make_taps(float gx, float gy) {
    // ((gx+1)*(Wp-1))*0.5 == (gx+1)*((Wp-1)*0.5) bit-exactly (x0.5 is exact)
    const float ixf = (gx + 1.0f) * ((float)(WP_ - 1) * 0.5f);
    const float iyf = (gy + 1.0f) * ((float)(HP_ - 1) * 0.5f);
    const float x0f = floorf(ixf), y0f = floorf(iyf);
    const float wx1 = ixf - x0f,   wy1 = iyf - y0f;
    const float wx0 = 1.0f - wx1,  wy0 = 1.0f - wy1;
    const float x1f = x0f + 1.0f,  y1f = y0f + 1.0f;

    const bool vx0 = (x0f >= 0.0f) && (x0f < (float)WP_);
    const bool vx1 = (x1f >= 0.0f) && (x1f < (float)WP_);
    const bool vy0 = (y0f >= 0.0f) && (y0f < (float)HP_);
    const bool vy1 = (y1f >= 0.0f) && (y1f < (float)HP_);

    const int xa = (int)fminf(fmaxf(x0f, 0.0f), (float)(WP_ - 1));
    const int xb = (int)fminf(fmaxf(x1f, 0.0f), (float)(WP_ - 1));
    const int ya = (int)fminf(fmaxf(y0f, 0.0f), (float)(HP_ - 1));
    const int yb = (int)fminf(fmaxf(y1f, 0.0f), (float)(HP_ - 1));

    Taps t;
    t.w00 = (vx0 && vy0) ? wx0 * wy0 : 0.0f;
    t.w01 = (vx1 && vy0) ? wx1 * wy0 : 0.0f;
    t.w10 = (vx0 && vy1) ? wx0 * wy1 : 0.0f;
    t.w11 = (vx1 && vy1) ? wx1 * wy1 : 0.0f;
    t.o00 = ya * WP_ + xa;  t.o01 = ya * WP_ + xb;
    t.o10 = yb * WP_ + xa;  t.o11 = yb * WP_ + xb;
    return t;
}

// ---------------------------------------------------------------------------
// Channel sweep shared by both sampling kernels.
// 8-channel blocks: polar stride 512KB and out stride 1MB both fit the signed
// 24-bit instruction offset (+-8MB), so the 8 loads x 4 taps reuse the same
// 4 VGPR tap offsets with immediate channel offsets; pointers bump once per
// block. 32 gather loads in flight per thread hides L2/HBM latency.
// Output is write-once -> non-temporal stores keep polar_feat resident in L2.
// ---------------------------------------------------------------------------
__device__ __forceinline__ void sample_channels(const float* __restrict__ pb,
                                                float* __restrict__ ob,
                                                const Taps t) {
    #pragma unroll
    for (int cb = 0; cb < C_ / 8; ++cb) {
        #pragma unroll
        for (int k = 0; k < 8; ++k) {
            const float* __restrict__ pc = pb + k * PPLANE_;
            const float acc = pc[t.o00] * t.w00 + pc[t.o01] * t.w01 +
                              pc[t.o10] * t.w10 + pc[t.o11] * t.w11;
            __builtin_nontemporal_store(acc, ob + k * OPLANE_);
        }
        pb += 8 * PPLANE_;
        ob += 8 * OPLANE_;
    }
}

// ---------------------------------------------------------------------------
// Main path: fused, fully-coalesced output walk.
// grid = (H*W/256, B); one thread per cart pixel, loops over 64 channels.
// Lane = consecutive x -> full-cacheline coalesced stores.
// ---------------------------------------------------------------------------
__global__ __launch_bounds__(256) void p2c_fused(const float* __restrict__ polar,
                                                 const float* __restrict__ ref,
                                                 const float* __restrict__ gxv,
                                                 const float* __restrict__ gyv,
                                                 const int*   __restrict__ map,
                                                 float* __restrict__ out) {
    const int p    = blockIdx.x * 256 + threadIdx.x;    // pixel id in [0, H*W)
    const int b    = blockIdx.y;
    const int base = b * (C_ * OPLANE_) + p;            // (b, c=0, y, x)

    const int n = map[p];
    if (n < 0) {
        // unmapped (~0.3% of pixels): copy ref_feat (read-once, write-once)
        const float* __restrict__ rb = ref + base;
        float* __restrict__ ob = out + base;
        #pragma unroll
        for (int cb = 0; cb < C_ / 8; ++cb) {
            #pragma unroll
            for (int k = 0; k < 8; ++k)
                __builtin_nontemporal_store(
                    __builtin_nontemporal_load(rb + k * OPLANE_),
                    ob + k * OPLANE_);
            rb += 8 * OPLANE_;
            ob += 8 * OPLANE_;
        }
        return;
    }

    const Taps t = make_taps(gxv[n], gyv[n]);
    sample_channels(polar + b * (C_ * PPLANE_), out + base, t);
}

// ---------------------------------------------------------------------------
// Fallback path (only if d_ws cannot hold the 1 MB map): copy + direct scatter
// ---------------------------------------------------------------------------
__global__ __launch_bounds__(256) void p2c_copy4(const v4f* __restrict__ src,
                                                 v4f* __restrict__ dst, int n4) {
    int i = blockIdx.x * 256 + threadIdx.x;
    if (i < n4)
        __builtin_nontemporal_store(__builtin_nontemporal_load(src + i), dst + i);
}

__global__ __launch_bounds__(256) void p2c_scatter(const float* __restrict__ polar,
                                                   const float* __restrict__ gxv,
                                                   const float* __restrict__ gyv,
                                                   const int*   __restrict__ iyv,
                                                   const int*   __restrict__ ixv,
                                                   int N, float* __restrict__ out) {
    const int n = blockIdx.x * 256 + threadIdx.x;
    const int b = blockIdx.y;
    if (n >= N) return;

    const int base = b * (C_ * OPLANE_) + iyv[n] * W_ + ixv[n];
    const Taps t = make_taps(gxv[n], gyv[n]);
    sample_channels(polar + b * (C_ * PPLANE_), out + base, t);
}

// ---------------------------------------------------------------------------
// Launcher
// inputs: [0] polar_feat f32[B,C,HP,WP]  [1] ref_feat f32[B,C,H,W]
//         [2] grid_x f32[N] [3] grid_y f32[N] [4] idx_y i32[N] [5] idx_x i32[N]
// output: f32[B,C,H,W]
// ---------------------------------------------------------------------------
extern "C" void kernel_launch(void* const* d_in, const int* in_sizes, int n_in,
                              void* d_out, int out_size, void* d_ws, size_t ws_size,
                              hipStream_t stream) {
    const float* polar = (const float*)d_in[0];
    const float* ref   = (const float*)d_in[1];
    const float* gx    = (const float*)d_in[2];
    const float* gy    = (const float*)d_in[3];
    const int*   iy    = (const int*)d_in[4];
    const int*   ix    = (const int*)d_in[5];
    float*       out   = (float*)d_out;
    const int N = in_sizes[2];

    if (ws_size >= (size_t)H_ * W_ * sizeof(int)) {
        int* map = (int*)d_ws;
        // 0xFF bytes == -1 per int; hipMemsetAsync is graph-capturable
        (void)hipMemsetAsync(map, 0xFF, (size_t)H_ * W_ * sizeof(int), stream);
        p2c_build_map<<<(N + 255) / 256, 256, 0, stream>>>(iy, ix, N, map);
        dim3 grid((H_ * W_) / 256, B_);
        p2c_fused<<<grid, 256, 0, stream>>>(polar, ref, gx, gy, map, out);
    } else {
        const int n4 = (B_ * C_ * H_ * W_) / 4;
        p2c_copy4<<<(n4 + 255) / 256, 256, 0, stream>>>((const v4f*)ref,
                                                        (v4f*)out, n4);
        dim3 grid((N + 255) / 256, B_);
        p2c_scatter<<<grid, 256, 0, stream>>>(polar, gx, gy, iy, ix, N, out);
    }
}